// Markov_50216757625127
// MI455X (gfx1250) — compile-verified
//
#include <hip/hip_runtime.h>
#include <stdint.h>

// Markov n-gram estimation: out[i] = count(pkey_i) / (count(ckey_i) + 1)
//   ckey_i = t[i-3]*1024^2 + t[i-2]*1024 + t[i-1]  (zero padded), pkey = ckey*1024 + t[i]
// Strategy: two open-addressing hash tables (2^23 slots each) in d_ws, counts
// accumulated with float atomics. Tables total 160 MB -> resident in 192 MB L2.
// Token tiles staged to LDS via the CDNA5 Tensor Data Mover where available.

#define NUM_WORDS   1024
#define ORDER       3
#define BLOCK       256
#define LOG2_SLOTS  23
#define SLOTS       (1u << LOG2_SLOTS)
#define HMASK       (SLOTS - 1u)
#define EMPTY32     0xFFFFFFFFu
#define EMPTY64     0xFFFFFFFFFFFFFFFFull

#if __has_builtin(__builtin_amdgcn_tensor_load_to_lds) && __has_builtin(__builtin_amdgcn_s_wait_tensorcnt)
#define USE_TDM 1
#else
#define USE_TDM 0
#endif

typedef uint32_t v4u __attribute__((ext_vector_type(4)));
typedef int      v4i __attribute__((ext_vector_type(4)));
typedef int      v8i __attribute__((ext_vector_type(8)));

__device__ __forceinline__ uint64_t mix64(uint64_t x) {
    x ^= x >> 30; x *= 0xbf58476d1ce4e5b9ull;
    x ^= x >> 27; x *= 0x94d049bb133111ebull;
    x ^= x >> 31;
    return x;
}

// Stage tokens[base-3 .. base+BLOCK-1] into toks[0 .. BLOCK+2] (zeros out of range).
__device__ __forceinline__ void load_tile(int* toks, const int* __restrict__ tokens,
                                          int base, int n) {
#if USE_TDM
    if (threadIdx.x < 32) {                    // wave 0 only: one TDM op per block
        if (threadIdx.x < ORDER) {             // 3-token halo via plain stores
            int g = base - ORDER + (int)threadIdx.x;
            toks[threadIdx.x] = (g >= 0) ? tokens[g] : 0;
        }
        // Build Tensor DMA Descriptor (D#): 1-D tile of `rem` dwords, OOB reads -> 0.
        uint32_t rem = (uint32_t)((n - base) < BLOCK ? (n - base) : BLOCK); // valid elems
        uint64_t ga  = (uint64_t)(uintptr_t)(tokens + base);
        uint32_t lds = (uint32_t)(uintptr_t)(&toks[ORDER]);  // low 32 bits = LDS byte offset
        v4u g0 = { 1u,                                       // count=1 valid descriptor
                   lds,                                      // lds_addr
                   (uint32_t)ga,                             // global_addr[31:0]
                   (uint32_t)((ga >> 32) & 0x1FFFFFFu) | (2u << 30) }; // ga[56:32] | type=2
        v8i g1 = { (int)(2u << 16),                          // wg_mask=0, data_size=4B
                   (int)((rem & 0xFFFFu) << 16),             // tensor_dim0[15:0]
                   (int)(((rem >> 16) & 0xFFFFu) | (1u << 16)), // tensor_dim0[31:16], tensor_dim1=1
                   (int)((uint32_t)BLOCK << 16),             // tile_dim0 = 256
                   0,                                        // tile_dim1/2 unused
                   (int)rem,                                 // tensor_dim0_stride[31:0] (unused, 1 row)
                   0, 0 };
        v4i g2 = { 0, 0, 0, 0 };
        v4i g3 = { 0, 0, 0, 0 };
#if defined(__clang_major__) && (__clang_major__ >= 23)
        v8i g4 = { 0, 0, 0, 0, 0, 0, 0, 0 };
        __builtin_amdgcn_tensor_load_to_lds(g0, g1, g2, g3, g4, 0);
#else
        __builtin_amdgcn_tensor_load_to_lds(g0, g1, g2, g3, 0);
#endif
        __builtin_amdgcn_s_wait_tensorcnt(0);
    }
    __syncthreads();
#else
    for (int t = (int)threadIdx.x; t < BLOCK + ORDER; t += BLOCK) {
        int g = base - ORDER + t;
        toks[t] = (g >= 0 && g < n) ? tokens[g] : 0;
    }
    __syncthreads();
#endif
}

__device__ __forceinline__ void insert32(uint32_t* keys, float* cnt, uint32_t key, float w) {
    uint32_t h = (uint32_t)mix64((uint64_t)key) & HMASK;
    for (;;) {
        uint32_t k = keys[h];                  // keys are write-once (EMPTY -> key)
        if (k == key) { atomicAdd(&cnt[h], w); return; }
        if (k == EMPTY32) {
            uint32_t prev = atomicCAS(&keys[h], EMPTY32, key);
            if (prev == EMPTY32 || prev == key) { atomicAdd(&cnt[h], w); return; }
        }
        h = (h + 1u) & HMASK;
    }
}

__device__ __forceinline__ void insert64(unsigned long long* keys, float* cnt,
                                         unsigned long long key, float w) {
    uint32_t h = (uint32_t)mix64(key) & HMASK;
    for (;;) {
        unsigned long long k = keys[h];
        if (k == key) { atomicAdd(&cnt[h], w); return; }
        if (k == EMPTY64) {
            unsigned long long prev = atomicCAS(&keys[h], EMPTY64, key);
            if (prev == EMPTY64 || prev == key) { atomicAdd(&cnt[h], w); return; }
        }
        h = (h + 1u) & HMASK;
    }
}

__device__ __forceinline__ float lookup32(const uint32_t* keys, const float* cnt, uint32_t key) {
    uint32_t h = (uint32_t)mix64((uint64_t)key) & HMASK;
    for (;;) {
        uint32_t k = keys[h];
        if (k == key) return cnt[h];
        if (k == EMPTY32) return 0.0f;         // unreachable: key was inserted in pass 1
        h = (h + 1u) & HMASK;
    }
}

__device__ __forceinline__ float lookup64(const unsigned long long* keys, const float* cnt,
                                          unsigned long long key) {
    uint32_t h = (uint32_t)mix64(key) & HMASK;
    for (;;) {
        unsigned long long k = keys[h];
        if (k == key) return cnt[h];
        if (k == EMPTY64) return 0.0f;
        h = (h + 1u) & HMASK;
    }
}

__device__ __forceinline__ void make_keys(const int* toks, uint32_t lt,
                                          uint32_t& ckey, unsigned long long& pkey) {
    uint32_t t3 = (uint32_t)toks[lt + 0];
    uint32_t t2 = (uint32_t)toks[lt + 1];
    uint32_t t1 = (uint32_t)toks[lt + 2];
    uint32_t t0 = (uint32_t)toks[lt + 3];
    ckey = (t3 * NUM_WORDS + t2) * NUM_WORDS + t1;              // < 2^30
    pkey = (unsigned long long)ckey * NUM_WORDS + t0;           // < 2^40
}

__global__ void __launch_bounds__(BLOCK)
markov_count(const int* __restrict__ tokens, const float* __restrict__ weights,
             uint32_t* __restrict__ cKeys, float* __restrict__ cCnt,
             unsigned long long* __restrict__ pKeys, float* __restrict__ pCnt, int n) {
    __shared__ int toks[BLOCK + ORDER + 1];
    int base = (int)blockIdx.x * BLOCK;
    load_tile(toks, tokens, base, n);

    int i = base + (int)threadIdx.x;
    if (i >= n) return;
    uint32_t ckey; unsigned long long pkey;
    make_keys(toks, threadIdx.x, ckey, pkey);
    float w = weights[i];
    insert32(cKeys, cCnt, ckey, w);
    insert64(pKeys, pCnt, pkey, w);
}

__global__ void __launch_bounds__(BLOCK)
markov_lookup(const int* __restrict__ tokens,
              const uint32_t* __restrict__ cKeys, const float* __restrict__ cCnt,
              const unsigned long long* __restrict__ pKeys, const float* __restrict__ pCnt,
              float* __restrict__ out, int n) {
    __shared__ int toks[BLOCK + ORDER + 1];
    int base = (int)blockIdx.x * BLOCK;
    load_tile(toks, tokens, base, n);

    int i = base + (int)threadIdx.x;
    if (i >= n) return;
    uint32_t ckey; unsigned long long pkey;
    make_keys(toks, threadIdx.x, ckey, pkey);
    float cc = lookup32(cKeys, cCnt, ckey);
    float cp = lookup64(pKeys, pCnt, pkey);
    out[i] = cp / (cc + 1.0f);
}

extern "C" void kernel_launch(void* const* d_in, const int* in_sizes, int n_in,
                              void* d_out, int out_size, void* d_ws, size_t ws_size,
                              hipStream_t stream) {
    const int*   tokens  = (const int*)d_in[0];
    const float* weights = (const float*)d_in[1];
    float*       out     = (float*)d_out;
    int n = in_sizes[0];

    // Workspace layout: 32MB ckey keys | 32MB ckey counts | 64MB pkey keys | 32MB pkey counts
    uint32_t*           cKeys = (uint32_t*)d_ws;
    float*              cCnt  = (float*)(cKeys + SLOTS);
    unsigned long long* pKeys = (unsigned long long*)(cCnt + SLOTS);
    float*              pCnt  = (float*)(pKeys + SLOTS);

    hipMemsetAsync(cKeys, 0xFF, (size_t)SLOTS * sizeof(uint32_t), stream);
    hipMemsetAsync(cCnt,  0x00, (size_t)SLOTS * sizeof(float), stream);
    hipMemsetAsync(pKeys, 0xFF, (size_t)SLOTS * sizeof(unsigned long long), stream);
    hipMemsetAsync(pCnt,  0x00, (size_t)SLOTS * sizeof(float), stream);

    int grid = (n + BLOCK - 1) / BLOCK;
    markov_count<<<grid, BLOCK, 0, stream>>>(tokens, weights, cKeys, cCnt, pKeys, pCnt, n);
    markov_lookup<<<grid, BLOCK, 0, stream>>>(tokens, cKeys, cCnt, pKeys, pCnt, out, n);
}